// DistributionLoss_87797721465214
// MI455X (gfx1250) — compile-verified
//
#include <hip/hip_runtime.h>
#include <math.h>

// ---------------------------------------------------------------------------
// Sliced-Wasserstein pyramid loss, MI455X (gfx1250, wave32).
// Stages (all on `stream`, graph-capture safe):
//   1) zero_out            : out[0] = 0
//   2) gen_dirs            : 8 levels x 64 unit directions (Box-Muller hash RNG)
//   3) sample_points       : fused bilinear-resize + subsample -> (img,lev,4096,4) f32
//   4) proj_wmma           : V_WMMA_F32_16X16X4_F32, D(16x16) = dirs(16x4) x pts(4x16)
//   5) sort_swd            : per (lev,dir) bitonic sort of both rows in LDS,
//                            mean((sorted_x - sorted_y)^2) -> atomic add into out
// ---------------------------------------------------------------------------

#define NLEV   8
#define NPROJ  64
#define MAXPTS 4096
#define HW     2048
#define BIGF   1.0e30f

typedef float v2f __attribute__((ext_vector_type(2)));
typedef float v8f __attribute__((ext_vector_type(8)));

__device__ __forceinline__ unsigned pcg(unsigned v) {
  unsigned s = v * 747796405u + 2891336453u;
  unsigned w = ((s >> ((s >> 28) + 4u)) ^ s) * 277803737u;
  return (w >> 22) ^ w;
}
__device__ __forceinline__ unsigned hash2(unsigned a, unsigned b) {
  return pcg(a * 0x9E3779B9u ^ pcg(b + 0x85EBCA6Bu));
}
__device__ __forceinline__ float u01(unsigned h) {
  return (float)(h >> 8) * (1.0f / 16777216.0f) + 1.0e-7f;
}

__global__ void zero_out_kernel(float* out) {
  if (threadIdx.x == 0) out[0] = 0.0f;
}

// 8 levels x 64 directions, each a normalized 3-vector, stored padded to 4 f32
// (4th component = 0 so the K=4 WMMA contributes nothing extra).
__global__ void gen_dirs_kernel(float* __restrict__ dirs) {
  int t = blockIdx.x * blockDim.x + threadIdx.x;
  if (t >= NLEV * NPROJ) return;
  int lev = t >> 6, d = t & 63;
  unsigned salt = 0xD1CE0001u + (unsigned)lev * 17u;
  float u1 = u01(hash2(salt, (unsigned)(d * 4 + 0)));
  float u2 = u01(hash2(salt, (unsigned)(d * 4 + 1)));
  float u3 = u01(hash2(salt, (unsigned)(d * 4 + 2)));
  float u4 = u01(hash2(salt, (unsigned)(d * 4 + 3)));
  float r1 = sqrtf(fmaxf(-2.0f * logf(u1), 0.0f));
  float z0 = r1 * cosf(6.28318530718f * u2);
  float z1 = r1 * sinf(6.28318530718f * u2);
  float z2 = sqrtf(fmaxf(-2.0f * logf(u3), 0.0f)) * cosf(6.28318530718f * u4);
  float inv = rsqrtf(z0 * z0 + z1 * z1 + z2 * z2 + 1.0e-20f);
  float* o = dirs + (size_t)t * 4;
  o[0] = z0 * inv; o[1] = z1 * inv; o[2] = z2 * inv; o[3] = 0.0f;
}

__device__ __forceinline__ void bilinear3(const float* __restrict__ img,
                                          float yc, float xc, float* o) {
  int y0 = (int)floorf(yc), x0 = (int)floorf(xc);
  int y1 = min(y0 + 1, HW - 1), x1 = min(x0 + 1, HW - 1);
  float wy = yc - (float)y0, wx = xc - (float)x0;
#pragma unroll
  for (int c = 0; c < 3; ++c) {
    const float* base = img + (size_t)c * HW * HW;
    __builtin_prefetch(base + (size_t)y1 * HW + x0, 0, 0);  // global_prefetch_b8
    float v00 = base[(size_t)y0 * HW + x0], v01 = base[(size_t)y0 * HW + x1];
    float v10 = base[(size_t)y1 * HW + x0], v11 = base[(size_t)y1 * HW + x1];
    o[c] = (1.0f - wy) * ((1.0f - wx) * v00 + wx * v01) +
           wy * ((1.0f - wx) * v10 + wx * v11);
  }
}

// Fused resize + subsample. Never materializes pyramid images: each sampled
// point bilinearly gathers its 4 full-res neighbors directly.
// pts layout: [(img*NLEV + lev)*MAXPTS + j] * 4  (xyz + 0 pad)
__global__ void sample_points_kernel(const float* __restrict__ warped,
                                     const float* __restrict__ target,
                                     float* __restrict__ pts) {
  int lev = blockIdx.x;
  int j = blockIdx.y * blockDim.x + threadIdx.x;   // 0..4095
  int s = 16 << lev;                               // 16,32,...,2048
  long long tot = (long long)s * (long long)s;
  int np = (tot < (long long)MAXPTS) ? (int)tot : MAXPTS;
  if (j >= np) return;
  float scale = (float)(HW - 1) / (float)(s - 1);  // align_corners=True
  const float* imgs[2] = {warped, target};
#pragma unroll
  for (int im = 0; im < 2; ++im) {
    int pix;
    if (tot <= (long long)MAXPTS) {
      pix = j;  // keep all pixels (matches reference for n <= MAX_POINTS)
    } else {    // deterministic hash subsample (different stream per image)
      pix = (int)(hash2(0x5A170000u + (unsigned)(lev * 2 + im), (unsigned)j) %
                  (unsigned long long)tot);
    }
    int oy = pix / s, ox = pix % s;
    float o[3];
    bilinear3(imgs[im], (float)oy * scale, (float)ox * scale, o);
    float* dst = pts + (((size_t)im * NLEV + lev) * MAXPTS + j) * 4;
    dst[0] = o[0]; dst[1] = o[1]; dst[2] = o[2]; dst[3] = 0.0f;
  }
}

// One wave per 16x16 output tile: D = A(16 dirs x 4) * B(4 x 16 pts) + 0.
// 2 img * 8 lev * 4 dir-tiles * 256 pt-tiles = 16384 waves, all full EXEC.
__global__ void proj_wmma_kernel(const float* __restrict__ dirs,
                                 const float* __restrict__ pts,
                                 float* __restrict__ proj) {
  int wave = blockIdx.x * (blockDim.x >> 5) + (threadIdx.x >> 5);
  int lane = threadIdx.x & 31;
  int pt_tile  = wave & 255;
  int rest     = wave >> 8;
  int dir_tile = rest & 3;
  rest >>= 2;
  int im  = rest & 1;
  int lev = rest >> 1;

  int mn = lane & 15;            // row of A tile / column of B tile
  int kb = (lane >> 4) << 1;     // K base: lanes 0-15 -> K{0,1}, 16-31 -> K{2,3}

  const float* A = dirs + ((size_t)lev * NPROJ + (size_t)dir_tile * 16) * 4;
  v2f a; a.x = A[mn * 4 + kb]; a.y = A[mn * 4 + kb + 1];

  const float* B = pts + (((size_t)im * NLEV + lev) * MAXPTS +
                          (size_t)pt_tile * 16) * 4;
  v2f b; b.x = B[mn * 4 + kb]; b.y = B[mn * 4 + kb + 1];

  v8f c = {0.f, 0.f, 0.f, 0.f, 0.f, 0.f, 0.f, 0.f};
  // 8 args: (neg_a, A, neg_b, B, c_mod, C, reuse_a, reuse_b)
  c = __builtin_amdgcn_wmma_f32_16x16x4_f32(false, a, false, b,
                                            (short)0, c, false, false);

  int s = 16 << lev;
  long long tot = (long long)s * (long long)s;
  int np = (tot < (long long)MAXPTS) ? (int)tot : MAXPTS;
  int pt = pt_tile * 16 + mn;
  float* row0 = proj + (((size_t)im * NLEV + lev) * NPROJ +
                        (size_t)dir_tile * 16) * MAXPTS;
#pragma unroll
  for (int r = 0; r < 8; ++r) {
    int m = r + ((lane >> 4) << 3);   // D layout: v[r] lanes0-15 -> M=r, 16-31 -> M=8+r
    row0[(size_t)m * MAXPTS + pt] = (pt < np) ? c[r] : BIGF;  // sentinel pads
  }
}

// One block per (level, direction): bitonic-sort both 4096-rows in LDS and
// accumulate mean squared diff. BIGF pads sort to matching tail slots -> 0.
__global__ void sort_swd_kernel(const float* __restrict__ proj,
                                float* __restrict__ out) {
  __shared__ float sx[MAXPTS];
  __shared__ float sy[MAXPTS];
  __shared__ float red[256];
  int lev = blockIdx.x >> 6;
  int d   = blockIdx.x & 63;
  const float* rx = proj + (((size_t)0 * NLEV + lev) * NPROJ + d) * MAXPTS;
  const float* ry = proj + (((size_t)1 * NLEV + lev) * NPROJ + d) * MAXPTS;
  for (int i = threadIdx.x; i < MAXPTS; i += blockDim.x) {
    sx[i] = rx[i];
    sy[i] = ry[i];
  }
  __syncthreads();
  for (int k = 2; k <= MAXPTS; k <<= 1) {
    for (int j = k >> 1; j > 0; j >>= 1) {
      for (int i = threadIdx.x; i < MAXPTS; i += blockDim.x) {
        int ixj = i ^ j;
        if (ixj > i) {
          bool up = ((i & k) == 0);
          float a = sx[i], bb = sx[ixj];
          if ((a > bb) == up) { sx[i] = bb; sx[ixj] = a; }
          float cc = sy[i], dd = sy[ixj];
          if ((cc > dd) == up) { sy[i] = dd; sy[ixj] = cc; }
        }
      }
      __syncthreads();
    }
  }
  float acc = 0.0f;
  for (int i = threadIdx.x; i < MAXPTS; i += blockDim.x) {
    float df = sx[i] - sy[i];
    acc += df * df;
  }
  red[threadIdx.x] = acc;
  __syncthreads();
  for (int off = blockDim.x >> 1; off > 0; off >>= 1) {
    if (threadIdx.x < off) red[threadIdx.x] += red[threadIdx.x + off];
    __syncthreads();
  }
  if (threadIdx.x == 0) {
    int s = 16 << lev;
    long long tot = (long long)s * (long long)s;
    int np = (tot < (long long)MAXPTS) ? (int)tot : MAXPTS;
    atomicAdd(out, red[0] / ((float)NPROJ * (float)np));
  }
}

extern "C" void kernel_launch(void* const* d_in, const int* in_sizes, int n_in,
                              void* d_out, int out_size, void* d_ws, size_t ws_size,
                              hipStream_t stream) {
  (void)in_sizes; (void)n_in; (void)out_size; (void)ws_size;
  const float* warped = (const float*)d_in[0];
  const float* target = (const float*)d_in[1];
  float* out = (float*)d_out;

  // Workspace layout (~17.06 MiB total):
  float* dirs = (float*)d_ws;                              //  8*64*4       =   8 KiB
  float* pts  = dirs + (size_t)NLEV * NPROJ * 4;           //  2*8*4096*4   =   1 MiB
  float* proj = pts + (size_t)2 * NLEV * MAXPTS * 4;       //  2*8*64*4096  =  16 MiB

  zero_out_kernel<<<1, 32, 0, stream>>>(out);
  gen_dirs_kernel<<<(NLEV * NPROJ + 255) / 256, 256, 0, stream>>>(dirs);
  sample_points_kernel<<<dim3(NLEV, MAXPTS / 256), 256, 0, stream>>>(warped, target, pts);
  proj_wmma_kernel<<<2048, 256, 0, stream>>>(dirs, pts, proj);   // 16384 waves
  sort_swd_kernel<<<NLEV * NPROJ, 256, 0, stream>>>(proj, out);  // 512 blocks
}